// AttentionDropout_80831284511093
// MI455X (gfx1250) — compile-verified
//
#include <hip/hip_runtime.h>

// ---------------------------------------------------------------------------
// Problem constants (from reference shapes): B=32, C=64, H=W=128, POOL=8
//   value = avgpool8(x)                                  -> [32, 16384]
//   A     = avgpool8((Wk x) * (Wq x))                    -> [32, 16384]  (softmax skipped: monotonic)
//   out[b, rank(j)] = 0 for j < BottomK or j > TopK; else out = value
// ---------------------------------------------------------------------------

typedef float v2f __attribute__((ext_vector_type(2)));
typedef float v8f __attribute__((ext_vector_type(8)));

#define NB    32
#define NC    64
#define NH    128
#define NW    128
#define NPOOL 16          // pooled grid 16x16
#define NROW  16384       // per-batch flattened pooled size = 64*16*16

// ---------------------------------------------------------------------------
// Kernel 1: value = avgpool8(x), written straight into d_out (full coverage).
// One wave per (b, c, ph): reads 8 rows x 128 floats, emits 16 pooled means.
// ---------------------------------------------------------------------------
__global__ __launch_bounds__(256) void value_kernel(const float* __restrict__ x,
                                                    float* __restrict__ out) {
    const int wid  = blockIdx.x * (blockDim.x >> 5) + (threadIdx.x >> 5);
    const int lane = threadIdx.x & 31;
    if (wid >= NB * NC * NPOOL) return;
    const int b  = wid / (NC * NPOOL);
    const int c  = (wid / NPOOL) % NC;
    const int ph = wid % NPOOL;

    const float* px = x + (((size_t)b * NC + c) * NH + ph * 8) * NW;
    float s = 0.0f;
#pragma unroll
    for (int r = 0; r < 8; ++r) {
        const float4 v = ((const float4*)(px + r * NW))[lane];   // cols 4*lane..4*lane+3
        s += v.x + v.y + v.z + v.w;
    }
    // lane covers 4 columns; pool column pw spans lanes 2*pw, 2*pw+1
    s += __shfl_xor(s, 1, 32);
    if ((lane & 1) == 0) {
        const int pw = lane >> 1;
        out[(size_t)b * NROW + c * 256 + ph * 16 + pw] = s * (1.0f / 64.0f);
    }
}

// ---------------------------------------------------------------------------
// Kernel 2: attention logits via f32 WMMA.
// One wave per 8x8 pool block (8 waves / workgroup). GEMM:
//   D[64o x 16pix] = W[64x64] * X[64 x 16pix] with V_WMMA_F32_16X16X4_F32;
// 4 N-tiles of 16 pixels (loop kept ROLLED to cap VGPR pressure < 256),
// 4 M-tiles, 16 K-steps. Pool sum over the 16 pixel-lanes of each half via
// shfl_xor butterfly (C/D layout: lanes 0-15 <-> M=r, lanes 16-31 <-> M=r+8).
// ---------------------------------------------------------------------------
__global__ __launch_bounds__(256) void logits_kernel(const float* __restrict__ x,
                                                     const float* __restrict__ Wk,
                                                     const float* __restrict__ Wq,
                                                     float* __restrict__ A_out) {
    __shared__ float lW[2 * NC * NC];          // Wk then Wq, 32 KB
    for (int i = threadIdx.x; i < NC * NC; i += blockDim.x) {
        lW[i]           = Wk[i];
        lW[NC * NC + i] = Wq[i];
    }
    __syncthreads();

    const int wave = threadIdx.x >> 5;
    const int lane = threadIdx.x & 31;
    const int blk  = blockIdx.x * 8 + wave;    // 0..8191 pool blocks
    const int b    = blk >> 8;
    const int ph   = (blk >> 4) & 15;
    const int pw   = blk & 15;

    const int n    = lane & 15;                // pixel column / A-matrix M within tile
    const int half = lane >> 4;                // 0: K+0/1, M=r ; 1: K+2/3, M=r+8

    v8f acc[4] = {v8f{}, v8f{}, v8f{}, v8f{}};

#pragma unroll 1   // keep rolled: one N-tile of B operands live at a time
    for (int t = 0; t < 4; ++t) {              // N-tile: pixel rows 2t, 2t+1
        const int row = 2 * t + (n >> 3);
        const int col = n & 7;
        const float* px = x + (size_t)b * (NC * NH * NW)
                            + (size_t)(ph * 8 + row) * NW + pw * 8 + col;
        // B-matrix tiles for all 16 K-steps: lane holds channels c0, c0+1 at pixel n
        v2f Bt[16];
#pragma unroll
        for (int k = 0; k < 16; ++k) {
            const int c0 = 4 * k + 2 * half;
            Bt[k].x = px[(size_t)c0 * (NH * NW)];
            Bt[k].y = px[(size_t)(c0 + 1) * (NH * NW)];
        }
#pragma unroll
        for (int mt = 0; mt < 4; ++mt) {
            v8f dk = v8f{};
            v8f dq = v8f{};
#pragma unroll
            for (int k = 0; k < 16; ++k) {
                const int c0   = 4 * k + 2 * half;
                const int wrow = (mt * 16 + n) * NC + c0;
                v2f ak, aq;
                ak.x = lW[wrow];            ak.y = lW[wrow + 1];
                aq.x = lW[NC * NC + wrow];  aq.y = lW[NC * NC + wrow + 1];
                dk = __builtin_amdgcn_wmma_f32_16x16x4_f32(
                        false, ak, false, Bt[k], (short)0, dk, false, false);
                dq = __builtin_amdgcn_wmma_f32_16x16x4_f32(
                        false, aq, false, Bt[k], (short)0, dq, false, false);
            }
            acc[mt] += dk * dq;             // key*query, elementwise per (o, pixel)
        }
    }

    // Reduce over the 16 pixel lanes inside each half; lane half holds M=r / M=r+8.
#pragma unroll
    for (int mt = 0; mt < 4; ++mt) {
        v8f v = acc[mt];
#pragma unroll
        for (int off = 1; off < 16; off <<= 1) {
#pragma unroll
            for (int r = 0; r < 8; ++r) v[r] += __shfl_xor(v[r], off, 32);
        }
        if (n == 0) {
            const int obase = mt * 16 + half * 8;
#pragma unroll
            for (int r = 0; r < 8; ++r) {
                const int o = obase + r;
                A_out[(size_t)b * NROW + o * 256 + ph * 16 + pw] = v[r] * (1.0f / 64.0f);
            }
        }
    }
}

// ---------------------------------------------------------------------------
// Kernel 3: for each "bad" original index j (j<BottomK or j>TopK), compute its
// stable descending rank by comparison counting against the LDS-resident row,
// then scatter out[b, rank] = 0. Ranks are a permutation -> deterministic.
// grid = (32 chunks, 32 batches), 256 threads, 1 j per thread.
// ---------------------------------------------------------------------------
__global__ __launch_bounds__(256) void rank_kernel(const float* __restrict__ A,
                                                   const float* __restrict__ dropout,
                                                   float* __restrict__ out) {
    __shared__ float row[NROW];               // 64 KB (LDS is 320 KB/WGP on CDNA5)
    const int b = blockIdx.y;
    for (int i = threadIdx.x; i < NROW; i += blockDim.x)
        row[i] = A[(size_t)b * NROW + i];
    __syncthreads();

    const float dr      = dropout[0];
    const int   TopK    = (int)((1.0f - dr * 0.5f) * (float)NROW);   // 12288
    const int   BottomK = (int)(dr * 0.5f * (float)NROW);            // 4096
    const int   nlow    = BottomK;            // j in [0, BottomK)
    const int   nhigh   = (NROW - 1) - TopK;  // j in (TopK, NROW-1]

    const int slot = blockIdx.x * blockDim.x + threadIdx.x;
    if (slot >= nlow + nhigh) return;
    const int j  = (slot < nlow) ? slot : (TopK + 1 + (slot - nlow));
    const float aj = row[j];

    int cnt = 0;
    for (int i = 0; i < NROW; i += 4) {       // lockstep lanes -> LDS broadcast
        const float4 v = *(const float4*)(&row[i]);
        cnt += (v.x > aj) || (v.x == aj && (i + 0) < j);
        cnt += (v.y > aj) || (v.y == aj && (i + 1) < j);
        cnt += (v.z > aj) || (v.z == aj && (i + 2) < j);
        cnt += (v.w > aj) || (v.w == aj && (i + 3) < j);
    }
    out[(size_t)b * NROW + cnt] = 0.0f;
}

// ---------------------------------------------------------------------------
extern "C" void kernel_launch(void* const* d_in, const int* in_sizes, int n_in,
                              void* d_out, int out_size, void* d_ws, size_t ws_size,
                              hipStream_t stream) {
    const float* x  = (const float*)d_in[0];
    const float* Wk = (const float*)d_in[1];
    const float* Wq = (const float*)d_in[2];
    const float* dr = (const float*)d_in[3];
    float* out = (float*)d_out;
    float* A   = (float*)d_ws;                 // 32*16384*4 = 2 MB of scratch

    // 1) out = avgpool8(x) (writes every output element each call)
    value_kernel<<<(NB * NC * NPOOL) / 8, 256, 0, stream>>>(x, out);
    // 2) attention logits via f32 WMMA (8 pool blocks per workgroup)
    logits_kernel<<<(NB * NPOOL * NPOOL) / 8, 256, 0, stream>>>(x, Wk, Wq, A);
    // 3) zero-scatter by descending rank of bad indices
    rank_kernel<<<dim3(32, NB), 256, 0, stream>>>(A, dr, out);
}